// FKRM_85839216378385
// MI455X (gfx1250) — compile-verified
//
#include <hip/hip_runtime.h>
#include <math.h>

#define IMG    96
#define HW     9216        // 96*96
#define BATCH  2
#define NTOK   18432       // B*HW
#define NEMB   8192
#define NTILE  1152        // NTOK/16

typedef float v2f __attribute__((ext_vector_type(2)));
typedef float v8f __attribute__((ext_vector_type(8)));

// ---- workspace layout (float offsets) ----
#define WS_K     0u          // 8192*4  padded keys   (k0,k1,k2,0)
#define WS_V     32768u      // 8192*4  padded values
#define WS_Q     65536u      // 18432*4 padded, pre-scaled queries
#define WS_EE    139264u     // B*3*HW
#define WS_HH    194560u     // B*3*HW
#define WS_S     249856u     // B*HW   raw cosine similarity
#define WS_AO    268288u     // 18432*4 attention output (3 used)
#define WS_BMIN  342016u     // 72 per-block minima
#define WS_BMAX  342096u     // 72 per-block maxima
#define WS_SMIN  342176u     // 2
#define WS_SMAX  342180u     // 2

__device__ __forceinline__ float gelu_tanh(float x) {
    float x3 = x * x * x;
    float t  = tanhf(0.7978845608028654f * (x + 0.044715f * x3));
    return 0.5f * x * (1.0f + t);
}

__device__ __forceinline__ void ln3(float* x, const float* g, const float* b) {
    float m  = (x[0] + x[1] + x[2]) * (1.0f / 3.0f);
    float d0 = x[0] - m, d1 = x[1] - m, d2 = x[2] - m;
    float v  = (d0 * d0 + d1 * d1 + d2 * d2) * (1.0f / 3.0f);
    float iv = rsqrtf(v + 1e-5f);
    x[0] = d0 * iv * g[0] + b[0];
    x[1] = d1 * iv * g[1] + b[1];
    x[2] = d2 * iv * g[2] + b[2];
}

// ---------------------------------------------------------------------------
// Stage 1: codebook -> padded K/V; front pixels -> mlp_in -> LN -> scaled Q
// ---------------------------------------------------------------------------
__global__ void prep_kernel(const float* __restrict__ front,
                            const float* __restrict__ bg,
                            const float* __restrict__ kw,
                            const float* __restrict__ vw,
                            const float* __restrict__ miw1, const float* __restrict__ mib1,
                            const float* __restrict__ miw2, const float* __restrict__ mib2,
                            const float* __restrict__ n1g,  const float* __restrict__ n1b,
                            const float* __restrict__ qw,
                            float* __restrict__ ws) {
    int i = blockIdx.x * blockDim.x + threadIdx.x;
    if (i < NEMB) {
        float c0 = bg[i], c1 = bg[NEMB + i], c2 = bg[2 * NEMB + i];
        float* K = ws + WS_K + (size_t)i * 4;
        float* V = ws + WS_V + (size_t)i * 4;
#pragma unroll
        for (int j = 0; j < 3; ++j) {
            K[j] = kw[j * 3 + 0] * c0 + kw[j * 3 + 1] * c1 + kw[j * 3 + 2] * c2;
            V[j] = vw[j * 3 + 0] * c0 + vw[j * 3 + 1] * c1 + vw[j * 3 + 2] * c2;
        }
        K[3] = 0.0f;
        V[3] = 0.0f;
    }
    if (i < NTOK) {
        int b = i / HW, hw = i % HW;
        const float* fp = front + (size_t)b * 3 * HW + hw;
        float p0 = fp[0], p1 = fp[HW], p2 = fp[2 * HW];
        float h[6];
#pragma unroll
        for (int j = 0; j < 6; ++j)
            h[j] = gelu_tanh(miw1[j * 3 + 0] * p0 + miw1[j * 3 + 1] * p1 +
                             miw1[j * 3 + 2] * p2 + mib1[j]);
        float x[3];
#pragma unroll
        for (int d = 0; d < 3; ++d) {
            float a = mib2[d];
#pragma unroll
            for (int j = 0; j < 6; ++j) a += miw2[d * 6 + j] * h[j];
            x[d] = a;
        }
        ln3(x, n1g, n1b);
        const float scale = 0.57735026918962576f;  // 3^-0.5
        float* Q = ws + WS_Q + (size_t)i * 4;
#pragma unroll
        for (int j = 0; j < 3; ++j)
            Q[j] = scale * (qw[j * 3 + 0] * x[0] + qw[j * 3 + 1] * x[1] + qw[j * 3 + 2] * x[2]);
        Q[3] = 0.0f;
    }
}

// ---------------------------------------------------------------------------
// Stage 2: PSF path — 7x7 box stats via LDS halo tiles, 1x1 convs, cosine S
// grid (6,6,B), block 256 (16x16 pixels)
// ---------------------------------------------------------------------------
__global__ void psf_kernel(const float* __restrict__ front,
                           const float* __restrict__ back,
                           const float* __restrict__ ewp, const float* __restrict__ ebp,
                           const float* __restrict__ fwp, const float* __restrict__ fbp,
                           const float* __restrict__ gwp, const float* __restrict__ gbp,
                           const float* __restrict__ hwp, const float* __restrict__ hbp,
                           float* __restrict__ ws) {
    __shared__ float sF[22 * 22];
    __shared__ float sB[22 * 22];
    __shared__ float redmin[256];
    __shared__ float redmax[256];

    int tid = threadIdx.x;
    int tx = tid & 15, ty = tid >> 4;
    int b = blockIdx.z;
    int gx = blockIdx.x * 16 + tx;
    int gy = blockIdx.y * 16 + ty;

    float xn[3], gn[3], ad[3], bv[3];

    for (int c = 0; c < 3; ++c) {
        const float* fpl = front + (size_t)(b * 3 + c) * HW;
        const float* bpl = back  + (size_t)(b * 3 + c) * HW;
        for (int i = tid; i < 22 * 22; i += 256) {
            int ly = i / 22, lx = i % 22;
            int yy = (int)blockIdx.y * 16 + ly - 3;
            int xx = (int)blockIdx.x * 16 + lx - 3;
            bool in = (yy >= 0) && (yy < IMG) && (xx >= 0) && (xx < IMG);
            sF[i] = in ? fpl[yy * IMG + xx] : 0.0f;
            sB[i] = in ? bpl[yy * IMG + xx] : 0.0f;
        }
        __syncthreads();
        float s1f = 0.f, s2f = 0.f, s1b = 0.f, s2b = 0.f;
#pragma unroll
        for (int dy = 0; dy < 7; ++dy)
#pragma unroll
            for (int dx = 0; dx < 7; ++dx) {
                float fv = sF[(ty + dy) * 22 + tx + dx];
                float bb = sB[(ty + dy) * 22 + tx + dx];
                s1f += fv; s2f += fv * fv;
                s1b += bb; s2b += bb * bb;
            }
        float fv  = sF[(ty + 3) * 22 + tx + 3];
        float bvv = sB[(ty + 3) * 22 + tx + 3];
        const float invN = 1.0f / 49.0f;
        float mF = s1f * invN, mB = s1b * invN;
        float vF = (s2f - s1f * s1f * invN) * (1.0f / 48.0f);
        float vB = (s2b - s1b * s1b * invN) * (1.0f / 48.0f);
        float sdF = sqrtf(fmaxf(vF, 0.0f));
        float sdB = sqrtf(fmaxf(vB, 0.0f));
        xn[c] = (fv  - mF) / (sdF + 1e-8f);
        gn[c] = (bvv - mB) / (sdB + 1e-8f);
        ad[c] = xn[c] * sdB + mB;   // patch adain (no eps on y_std)
        bv[c] = bvv;
        __syncthreads();
    }

    float EE[3], FF[3], GG[3], HH[3];
#pragma unroll
    for (int o = 0; o < 3; ++o) {
        EE[o] = ewp[o * 3 + 0] * ad[0] + ewp[o * 3 + 1] * ad[1] + ewp[o * 3 + 2] * ad[2] + ebp[o];
        FF[o] = fwp[o * 3 + 0] * xn[0] + fwp[o * 3 + 1] * xn[1] + fwp[o * 3 + 2] * xn[2] + fbp[o];
        GG[o] = gwp[o * 3 + 0] * gn[0] + gwp[o * 3 + 1] * gn[1] + gwp[o * 3 + 2] * gn[2] + gbp[o];
        HH[o] = hwp[o * 3 + 0] * bv[0] + hwp[o * 3 + 1] * bv[1] + hwp[o * 3 + 2] * bv[2] + hbp[o];
    }
    float dot = 0.f, nf = 0.f, ng = 0.f;
#pragma unroll
    for (int c = 0; c < 3; ++c) {
        dot += FF[c] * GG[c];
        nf  += FF[c] * FF[c];
        ng  += GG[c] * GG[c];
    }
    float S = dot / (sqrtf(nf) * sqrtf(ng));

    int pix = gy * IMG + gx;
#pragma unroll
    for (int c = 0; c < 3; ++c) {
        ws[WS_EE + (size_t)(b * 3 + c) * HW + pix] = EE[c];
        ws[WS_HH + (size_t)(b * 3 + c) * HW + pix] = HH[c];
    }
    ws[WS_S + (size_t)b * HW + pix] = S;

    redmin[tid] = S;
    redmax[tid] = S;
    __syncthreads();
    for (int st = 128; st > 0; st >>= 1) {
        if (tid < st) {
            redmin[tid] = fminf(redmin[tid], redmin[tid + st]);
            redmax[tid] = fmaxf(redmax[tid], redmax[tid + st]);
        }
        __syncthreads();
    }
    if (tid == 0) {
        int blk = (int)blockIdx.y * 6 + (int)blockIdx.x;
        ws[WS_BMIN + b * 36 + blk] = redmin[0];
        ws[WS_BMAX + b * 36 + blk] = redmax[0];
    }
}

// ---------------------------------------------------------------------------
// Stage 3: per-batch min/max over the 36 block results
// ---------------------------------------------------------------------------
__global__ void minmax_kernel(float* __restrict__ ws) {
    __shared__ float smn[64];
    __shared__ float smx[64];
    int b = blockIdx.x;
    int t = threadIdx.x;
    float mn = INFINITY, mx = -INFINITY;
    if (t < 36) {
        mn = ws[WS_BMIN + b * 36 + t];
        mx = ws[WS_BMAX + b * 36 + t];
    }
    smn[t] = mn; smx[t] = mx;
    __syncthreads();
    for (int st = 32; st > 0; st >>= 1) {
        if (t < st) {
            smn[t] = fminf(smn[t], smn[t + st]);
            smx[t] = fmaxf(smx[t], smx[t + st]);
        }
        __syncthreads();
    }
    if (t == 0) {
        ws[WS_SMIN + b] = smn[0];
        ws[WS_SMAX + b] = smx[0];
    }
}

// ---------------------------------------------------------------------------
// Stage 4: WMMA attention, lane-local online softmax (zero per-tile cross-lane
// traffic). One wave = 16 tokens; 512 key-tiles via V_WMMA_F32_16X16X4_F32.
// C/D layout: reg r -> row r (lanes 0-15) / row 8+r (lanes 16-31), col = lane&15.
// Lane-partial (m,s,acc) softmax states are merged once at the end with the
// associative flash-merge, via a 4-step butterfly per 16-lane half.
// ---------------------------------------------------------------------------
__global__ __launch_bounds__(256, 2) void attn_kernel(float* __restrict__ ws) {
    const float* __restrict__ Km = ws + WS_K;
    const float* __restrict__ Vm = ws + WS_V;
    const float* __restrict__ Qm = ws + WS_Q;
    float* __restrict__ Ao = ws + WS_AO;

    int lane = threadIdx.x & 31;
    int wave = threadIdx.x >> 5;
    int tile = (int)blockIdx.x * 8 + wave;
    int t0 = tile * 16;
    int sub = lane & 15;
    int k0 = (lane < 16) ? 0 : 2;   // A/B hold K={0,1} in half0, K={2,3} in half1

    // A fragment: 16x4 Q tile (pre-scaled; lane holds {q[k0],q[k0+1]} of its row)
    v2f a = *(const v2f*)(Qm + (size_t)(t0 + sub) * 4 + k0);

    float m[8], s[8], ax[8], ay[8], az[8];
#pragma unroll
    for (int r = 0; r < 8; ++r) {
        m[r] = -INFINITY; s[r] = 0.f; ax[r] = 0.f; ay[r] = 0.f; az[r] = 0.f;
    }

#pragma unroll 2
    for (int nt = 0; nt < NEMB / 16; ++nt) {
        int n = nt * 16 + sub;
        v2f bq = *(const v2f*)(Km + (size_t)n * 4 + k0);
        float4 vv = *(const float4*)(Vm + (size_t)n * 4);
        __builtin_prefetch(Km + (size_t)(n + 32) * 4 + k0, 0, 3);
        __builtin_prefetch(Vm + (size_t)(n + 32) * 4, 0, 3);

        v8f cz = {0.f, 0.f, 0.f, 0.f, 0.f, 0.f, 0.f, 0.f};
        v8f lg = __builtin_amdgcn_wmma_f32_16x16x4_f32(
            false, a, false, bq, (short)0, cz, false, false);

        // lane-local online softmax update (no cross-lane ops in the hot loop)
#pragma unroll
        for (int r = 0; r < 8; ++r) {
            float l    = lg[r];
            float nm   = fmaxf(m[r], l);
            float corr = __expf(m[r] - nm);
            float p    = __expf(l - nm);
            s[r]  = s[r]  * corr + p;
            ax[r] = ax[r] * corr + p * vv.x;
            ay[r] = ay[r] * corr + p * vv.y;
            az[r] = az[r] * corr + p * vv.z;
            m[r] = nm;
        }
    }

    // merge lane-partial softmax states across each 16-lane half (once)
#pragma unroll
    for (int r = 0; r < 8; ++r) {
#pragma unroll
        for (int mk = 1; mk < 16; mk <<= 1) {
            float mo  = __shfl_xor(m[r],  mk);
            float so  = __shfl_xor(s[r],  mk);
            float axo = __shfl_xor(ax[r], mk);
            float ayo = __shfl_xor(ay[r], mk);
            float azo = __shfl_xor(az[r], mk);
            float M  = fmaxf(m[r], mo);
            float e1 = __expf(m[r] - M);
            float e2 = __expf(mo - M);
            s[r]  = s[r]  * e1 + so  * e2;
            ax[r] = ax[r] * e1 + axo * e2;
            ay[r] = ay[r] * e1 + ayo * e2;
            az[r] = az[r] * e1 + azo * e2;
            m[r] = M;
        }
    }
#pragma unroll
    for (int r = 0; r < 8; ++r) {
        if (sub == r) {
            int rowg = (lane < 16) ? r : (8 + r);
            size_t t = (size_t)(t0 + rowg) * 4;
            float inv = 1.0f / s[r];
            Ao[t + 0] = ax[r] * inv;
            Ao[t + 1] = ay[r] * inv;
            Ao[t + 2] = az[r] * inv;
            Ao[t + 3] = 0.0f;
        }
    }
}

// ---------------------------------------------------------------------------
// Stage 5: mlp_out -> LN2 -> min/max-normalized blend with PSF -> fuse conv
// ---------------------------------------------------------------------------
__global__ void finalize_kernel(const float* __restrict__ mow1, const float* __restrict__ mob1,
                                const float* __restrict__ mow2, const float* __restrict__ mob2,
                                const float* __restrict__ n2g,  const float* __restrict__ n2b,
                                const float* __restrict__ fusw, const float* __restrict__ fusb,
                                const float* __restrict__ ws,
                                float* __restrict__ out) {
    int t = blockIdx.x * blockDim.x + threadIdx.x;
    if (t >= NTOK) return;
    int b = t / HW, hw = t % HW;

    const float* A = ws + WS_AO + (size_t)t * 4;
    float x0 = A[0], x1 = A[1], x2 = A[2];
    float h[6];
#pragma unroll
    for (int j = 0; j < 6; ++j)
        h[j] = gelu_tanh(mow1[j * 3 + 0] * x0 + mow1[j * 3 + 1] * x1 +
                         mow1[j * 3 + 2] * x2 + mob1[j]);
    float y[3];
#pragma unroll
    for (int d = 0; d < 3; ++d) {
        float acc = mob2[d];
#pragma unroll
        for (int j = 0; j < 6; ++j) acc += mow2[d * 6 + j] * h[j];
        y[d] = acc;
    }
    ln3(y, n2g, n2b);

    float Sv   = ws[WS_S + t];
    float smin = ws[WS_SMIN + b];
    float smax = ws[WS_SMAX + b];
    float Sn = (Sv - smin) / (smax - smin);

    float fused[3];
#pragma unroll
    for (int c = 0; c < 3; ++c) {
        float ee = ws[WS_EE + (size_t)(b * 3 + c) * HW + hw];
        float hh = ws[WS_HH + (size_t)(b * 3 + c) * HW + hw];
        fused[c] = Sn * ee + (1.0f - Sn) * hh;
    }
#pragma unroll
    for (int o = 0; o < 3; ++o) {
        float acc = fusb[o];
#pragma unroll
        for (int i = 0; i < 3; ++i) acc += fusw[o * 6 + i] * y[i];
#pragma unroll
        for (int i = 0; i < 3; ++i) acc += fusw[o * 6 + 3 + i] * fused[i];
        out[(size_t)(b * 3 + o) * HW + hw] = acc;
    }
}

// ---------------------------------------------------------------------------
extern "C" void kernel_launch(void* const* d_in, const int* in_sizes, int n_in,
                              void* d_out, int out_size, void* d_ws, size_t ws_size,
                              hipStream_t stream) {
    const float* front  = (const float*)d_in[0];
    const float* back   = (const float*)d_in[1];
    // d_in[2] mask: unused by reference
    const float* bg     = (const float*)d_in[3];
    const float* mi_w1  = (const float*)d_in[4];
    const float* mi_b1  = (const float*)d_in[5];
    const float* mi_w2  = (const float*)d_in[6];
    const float* mi_b2  = (const float*)d_in[7];
    const float* q_w    = (const float*)d_in[8];
    const float* k_w    = (const float*)d_in[9];
    const float* v_w    = (const float*)d_in[10];
    const float* mo_w1  = (const float*)d_in[11];
    const float* mo_b1  = (const float*)d_in[12];
    const float* mo_w2  = (const float*)d_in[13];
    const float* mo_b2  = (const float*)d_in[14];
    const float* n1_g   = (const float*)d_in[15];
    const float* n1_b   = (const float*)d_in[16];
    const float* n2_g   = (const float*)d_in[17];
    const float* n2_b   = (const float*)d_in[18];
    const float* e_w    = (const float*)d_in[19];
    const float* e_b    = (const float*)d_in[20];
    const float* f_w    = (const float*)d_in[21];
    const float* f_b    = (const float*)d_in[22];
    const float* g_w    = (const float*)d_in[23];
    const float* g_b    = (const float*)d_in[24];
    const float* h_w    = (const float*)d_in[25];
    const float* h_b    = (const float*)d_in[26];
    const float* fuse_w = (const float*)d_in[27];
    const float* fuse_b = (const float*)d_in[28];

    float* ws  = (float*)d_ws;
    float* out = (float*)d_out;

    prep_kernel<<<(NTOK + 255) / 256, 256, 0, stream>>>(
        front, bg, k_w, v_w, mi_w1, mi_b1, mi_w2, mi_b2, n1_g, n1_b, q_w, ws);

    psf_kernel<<<dim3(6, 6, BATCH), 256, 0, stream>>>(
        front, back, e_w, e_b, f_w, f_b, g_w, g_b, h_w, h_b, ws);

    minmax_kernel<<<BATCH, 64, 0, stream>>>(ws);

    attn_kernel<<<NTILE / 8, 256, 0, stream>>>(ws);

    finalize_kernel<<<(NTOK + 255) / 256, 256, 0, stream>>>(
        mo_w1, mo_b1, mo_w2, mo_b2, n2_g, n2_b, fuse_w, fuse_b, ws, out);
}